// CubicSplineLayer_72395968741956
// MI455X (gfx1250) — compile-verified
//
#include <hip/hip_runtime.h>

// CubicSplineLayer for MI455X (gfx1250, wave32).
//
// out[i] = cm*FW[jm] + cp*FW[jp] + am*W[jm] + ap*W[jp] + (b - mean.W)
// computed as a 16x16x4 fp32 WMMA per 16 points:
//   A (16x4): row m = (cm, cp, am, ap) of point m   [lanes 0-15 hold K=0,1; lanes 16-31 hold K=2,3]
//   B (4x16): column n = (FW[jm], FW[jp], W[jm], W[jp]) of point n
//   C       : bias broadcast, so D[m, m] is the final answer for point m.
// Bandwidth-bound: 16 MB in + 16 MB out -> ~1.4 us floor @ 23.3 TB/s.

typedef __attribute__((ext_vector_type(2))) float v2f;
typedef __attribute__((ext_vector_type(8))) float v8f;

#define KNOTS 10

__device__ __forceinline__ v8f tile_wmma(
    float xv, bool hi, const float* kn,
    float k0, float kL, float h0, float hL, float rh0, float rhL, float bias,
    const float4* sIV, const float* sGW)
{
    // searchsorted(knots, x, side='left') = #knots strictly < x, clipped to [1, K-1]
    int ss = 0;
#pragma unroll
    for (int i = 0; i < KNOTS; ++i) ss += (kn[i] < xv) ? 1 : 0;
    const int j = ss < 1 ? 1 : (ss > KNOTS - 1 ? KNOTS - 1 : ss);

    const bool below = xv < k0;
    const bool above = xv > kL;

    const float4 iv = sIV[j];            // (knots[j-1], knots[j], 1/h, 0) : one ds_load_b128
    const float dm = iv.y - xv;
    const float dp = xv - iv.x;
    const float h  = iv.y - iv.x;
    const float rh = iv.z;
    const float am_i = dm * rh;
    const float ap_i = dp * rh;
    const float cm_i = (dm * dm * dm * rh - h * dm) * (1.0f / 6.0f);
    const float cp_i = (dp * dp * dp * rh - h * dp) * (1.0f / 6.0f);

    const float xb = xv - k0;
    const float xa = xv - kL;

    const float cm = below ? (-xb * h0 * (1.0f / 3.0f)) : (above ? (xa * hL * (1.0f / 6.0f)) : cm_i);
    const float cp = below ? (-xb * h0 * (1.0f / 6.0f)) : (above ? (xa * hL * (1.0f / 3.0f)) : cp_i);
    const float am = below ? (1.0f - xb * rh0)          : (above ? (-xa * rhL)               : am_i);
    const float ap = below ? (xb * rh0)                 : (above ? (1.0f + xa * rhL)         : ap_i);

    const int jm = below ? 0 : (above ? (KNOTS - 2) : (j - 1));
    // Fused gather table: [0..15] = FW (low half needs K=0,1), [16..31] = W (high half, K=2,3).
    // FW[0] == FW[K-1] == 0 (F's first/last rows are zero), matching the reference exactly.
    const int g = (hi ? 16 : 0) + jm;
    const float wm = sGW[g];
    const float wp = sGW[g + 1];         // one ds_load_2addr_b32 pair

    v2f a;  a.x = hi ? am : cm;  a.y = hi ? ap : cp;
    v2f bv; bv.x = wm;           bv.y = wp;
    v8f c = {bias, bias, bias, bias, bias, bias, bias, bias};
    return __builtin_amdgcn_wmma_f32_16x16x4_f32(
        /*neg_a=*/false, a, /*neg_b=*/false, bv,
        /*c_mod=*/(short)0, c, /*reuse_a=*/false, /*reuse_b=*/false);
}

__global__ __launch_bounds__(256) void spline_wmma_kernel(
    const float* __restrict__ x,
    const float* __restrict__ knots,
    const float* __restrict__ F,     // KNOTS x KNOTS
    const float* __restrict__ W,     // 1 x KNOTS
    const float* __restrict__ bptr,  // 1
    const float* __restrict__ mean,  // 1 x KNOTS
    float* __restrict__ out,
    int n)
{
    __shared__ float4 sIV[16];  // (knots[j-1], knots[j], 1/h_j, 0)
    __shared__ float  sGW[32];  // [0..15]=F@W^T, [16..31]=W

    const int t = threadIdx.x;
    if (t < 16) {
        float4 iv = make_float4(0.f, 0.f, 0.f, 0.f);
        if (t >= 1 && t < KNOTS) {
            iv.x = knots[t - 1];
            iv.y = knots[t];
            iv.z = 1.0f / (iv.y - iv.x);
        }
        sIV[t] = iv;
        float acc = 0.0f;
        if (t < KNOTS) {
#pragma unroll
            for (int k = 0; k < KNOTS; ++k) acc += F[t * KNOTS + k] * W[k];
        }
        sGW[t]      = acc;
        sGW[t + 16] = (t < KNOTS) ? W[t] : 0.0f;
    }
    __syncthreads();

    // Wave-uniform scalar constants (scalarized loads).
    float kn[KNOTS];
#pragma unroll
    for (int i = 0; i < KNOTS; ++i) kn[i] = knots[i];
    const float k0 = kn[0];
    const float kL = kn[KNOTS - 1];
    const float h0 = kn[1] - k0;
    const float hL = kL - kn[KNOTS - 2];
    const float rh0 = 1.0f / h0;
    const float rhL = 1.0f / hL;

    float meanW = 0.0f;
#pragma unroll
    for (int k = 0; k < KNOTS; ++k) meanW += mean[k] * W[k];
    const float bias = bptr[0] - meanW;

    const int lane = t & 31;
    const int sub  = lane & 15;    // point-within-tile (rows of the WMMA)
    const bool hi  = lane >= 16;   // high half supplies K=2,3

    const int fullTiles = n >> 4;  // complete 16-point tiles: unguarded hot loop
    const int waveId = blockIdx.x * 8 + (t >> 5);
    const int nWaves = gridDim.x * 8;

    for (int tile = waveId; tile < fullTiles; tile += nWaves) {
        const int p = (tile << 4) + sub;
        const float xv = x[p];
        // Stream-ahead prefetch of the next grid-stride tile (global_prefetch_b8).
        __builtin_prefetch(x + p + (nWaves << 4), 0, 1);

        v8f c = tile_wmma(xv, hi, kn, k0, kL, h0, hL, rh0, rhL, bias, sIV, sGW);

        // D rows 0..7 live in lane 0's c[0..7], rows 8..15 in lane 16's c[0..7]:
        // two B128 stores -> 64B contiguous per tile.
        if (sub == 0) {
            const int base = (tile << 4) + (hi ? 8 : 0);
            float4 lo4 = make_float4(c[0], c[1], c[2], c[3]);
            float4 hi4 = make_float4(c[4], c[5], c[6], c[7]);
            *(float4*)(out + base)     = lo4;
            *(float4*)(out + base + 4) = hi4;
        }
    }

    // Partial tail tile (n % 16 points): one designated wave, guarded stores.
    if ((n & 15) && waveId == 0) {
        const int tbase = fullTiles << 4;
        const int p = tbase + sub;
        const int pc = (p < n) ? p : (n - 1);  // clamp load: keep EXEC full for WMMA
        const float xv = x[pc];

        v8f c = tile_wmma(xv, hi, kn, k0, kL, h0, hL, rh0, rhL, bias, sIV, sGW);

        if (sub == 0) {
            const int base = tbase + (hi ? 8 : 0);
#pragma unroll
            for (int v = 0; v < 8; ++v)
                if (base + v < n) out[base + v] = c[v];
        }
    }
}

extern "C" void kernel_launch(void* const* d_in, const int* in_sizes, int n_in,
                              void* d_out, int out_size, void* d_ws, size_t ws_size,
                              hipStream_t stream) {
    const float* x     = (const float*)d_in[0];
    const float* knots = (const float*)d_in[1];
    const float* F     = (const float*)d_in[2];
    const float* W     = (const float*)d_in[3];
    const float* b     = (const float*)d_in[4];
    const float* mean  = (const float*)d_in[5];
    float* out = (float*)d_out;

    const int n = in_sizes[0];
    const long long nTiles = ((long long)n + 15) >> 4;
    int blocks = (int)((nTiles + 7) / 8);   // 8 waves (tiles) per 256-thread block
    if (blocks > 2048) blocks = 2048;       // grid-stride beyond this
    if (blocks < 1) blocks = 1;

    spline_wmma_kernel<<<blocks, 256, 0, stream>>>(x, knots, F, W, b, mean, out, n);
}